// LiteNTK_45561013076150
// MI455X (gfx1250) — compile-verified
//
#include <hip/hip_runtime.h>
#include <math.h>

typedef __attribute__((ext_vector_type(2))) float v2f;
typedef __attribute__((ext_vector_type(8))) float v8f;

#define PI_F     3.14159265358979323846f
#define INV_PI_F 0.31830988618379067154f
#define NN   64   // nodes per graph
#define KAUG 68   // 64 features + 1 rank-1 aug col + 3 zero pad (K % 4 == 0)

// ---------------------------------------------------------------------------
// Per-graph preprocessing: rowsum r, colsum c of A' = A + 1e-4 I, and
// Y1 = diag(1/r) A' X, Y2 = diag(1/c) A' X, augmented:
//   Y1[:,64] = 0.01, Y2[b,64] = 0.01 * r_b/c_b  (so dot gives 1e-4 * e1*e2)
// ---------------------------------------------------------------------------
__global__ void __launch_bounds__(256) prep_kernel(
    const float* __restrict__ A, const float* __restrict__ X,
    float* __restrict__ Y1, float* __restrict__ Y2,
    float* __restrict__ rbuf, float* __restrict__ cbuf) {
  int g = blockIdx.x;
  const float* Ag = A + (size_t)g * NN * NN;
  const float* Xg = X + (size_t)g * NN * NN;
  __shared__ float r[NN], c[NN];
  int tid = threadIdx.x;
  if (tid < NN) {
    float s = 1e-4f;                       // diag eps contribution
    for (int j = 0; j < NN; ++j) s += Ag[tid * NN + j];
    r[tid] = s;
  } else if (tid < 2 * NN) {
    int i = tid - NN;
    float s = 1e-4f;
    for (int j = 0; j < NN; ++j) s += Ag[j * NN + i];
    c[i] = s;
  }
  __syncthreads();
  if (tid < NN) rbuf[g * NN + tid] = r[tid];
  else if (tid < 2 * NN) cbuf[g * NN + (tid - NN)] = c[tid - NN];
  for (int e = tid; e < NN * KAUG; e += 256) {
    int a = e / KAUG, k = e - a * KAUG;
    float y1, y2;
    if (k < NN) {
      float num = 1e-4f * Xg[a * NN + k];  // (A' diagonal eps) * X[a,k]
      for (int j = 0; j < NN; ++j) num += Ag[a * NN + j] * Xg[j * NN + k];
      y1 = num / r[a];
      y2 = num / c[a];
    } else if (k == NN) {
      y1 = 0.01f;
      y2 = 0.01f * r[a] / c[a];
    } else { y1 = 0.f; y2 = 0.f; }
    Y1[((size_t)g * NN + a) * KAUG + k] = y1;
    Y2[((size_t)g * NN + a) * KAUG + k] = y2;
  }
}

// ---------------------------------------------------------------------------
// Per-graph: diag normalizers d1,d2 (layer 2 reduces to the clipped-ratio
// constant next(0.9999)) and reduction vectors p1 = colsum(A~1), p2 = colsum(A~2)
// ---------------------------------------------------------------------------
__global__ void __launch_bounds__(64) postprep_kernel(
    const float* __restrict__ A,
    const float* __restrict__ Y1, const float* __restrict__ Y2,
    const float* __restrict__ rbuf, const float* __restrict__ cbuf,
    float* __restrict__ dbuf, float* __restrict__ p1, float* __restrict__ p2) {
  int g = blockIdx.x;
  int i = threadIdx.x;
  const float* y1 = Y1 + ((size_t)g * NN + i) * KAUG;
  const float* y2 = Y2 + ((size_t)g * NN + i) * KAUG;
  float s = 0.f;
  for (int k = 0; k < KAUG; ++k) s += y1[k] * y2[k];
  float d1 = sqrtf(fmaxf(s, 0.f));
  // layer-2 diag: ratio == 1.0 clipped to 0.9999; apply the arccos update
  const float cv = 0.9999f;
  float ac = acosf(cv);
  float pm = PI_F - ac;
  float Sn = (cv * pm + sqrtf(1.f - cv * cv)) * INV_PI_F;
  float d2 = d1 * sqrtf(Sn);
  dbuf[((size_t)g * 2 + 0) * NN + i] = d1;
  dbuf[((size_t)g * 2 + 1) * NN + i] = d2;
  const float* Ag = A + (size_t)g * NN * NN;
  float s1 = 0.f, s2 = 0.f;
  for (int a = 0; a < NN; ++a) {
    float v = Ag[a * NN + i] + ((a == i) ? 1e-4f : 0.f);
    s1 += v / rbuf[g * NN + a];
    s2 += v / cbuf[g * NN + a];
  }
  p1[g * NN + i] = s1;
  p2[g * NN + i] = s2;
}

// ---------------------------------------------------------------------------
// NTK arc-cosine layer update (elementwise)
// ---------------------------------------------------------------------------
__device__ __forceinline__ void ntk_layer(float& sig, float& ntk, float dl, float dr) {
  float tmp = dl * dr + 1e-6f;
  float sv = sig / tmp;
  sv = fminf(fmaxf(sv, -0.9999f), 0.9999f);
  float ac = acosf(sv);
  float pm = PI_F - ac;
  float Sn = (sv * pm + sqrtf(fmaxf(1.f - sv * sv, 0.f))) * INV_PI_F;
  ntk = ntk * (pm * INV_PI_F) + Sn;
  sig = Sn * tmp;
}

// ---------------------------------------------------------------------------
// Pairwise NTK: one workgroup (4 waves) per graph pair.
// sigma(64x64) = Y1L[g1] * Y2R[g2]^T via V_WMMA_F32_16X16X4_F32 over K=68,
// then elementwise 2-layer recursion, then (1/4096) * p1^T ntk p2.
// ---------------------------------------------------------------------------
__global__ void __launch_bounds__(128) pairwise_kernel(
    const float* __restrict__ Y1L, const float* __restrict__ dLb, const float* __restrict__ pLb,
    const float* __restrict__ Y2R, const float* __restrict__ dRb, const float* __restrict__ pRb,
    float* __restrict__ out, int Mright) {
  int g2 = blockIdx.x, g1 = blockIdx.y;
  const float* y1 = Y1L + (size_t)g1 * NN * KAUG;
  const float* y2 = Y2R + (size_t)g2 * NN * KAUG;
  const float* dL = dLb + (size_t)g1 * 2 * NN;
  const float* dR = dRb + (size_t)g2 * 2 * NN;
  const float* pL = pLb + (size_t)g1 * NN;
  const float* pR = pRb + (size_t)g2 * NN;

  int lane   = threadIdx.x & 31;
  int wave   = threadIdx.x >> 5;   // 0..3 -> 16-row strip
  int laneLo = lane & 15;
  int hi     = lane >> 4;          // lane half selects K pair {2hi, 2hi+1}

  v8f z = {0.f, 0.f, 0.f, 0.f, 0.f, 0.f, 0.f, 0.f};
  v8f acc[4] = {z, z, z, z};

  // A frag: row = strip + laneLo, ks = k0 + 2*hi .. +1   (16x4 f32 layout)
  // B frag: col = tile*16 + laneLo, same k selection      (4x16 f32 layout)
  const float* aRow = y1 + (wave * 16 + laneLo) * KAUG + 2 * hi;
  const float* bRow = y2 + laneLo * KAUG + 2 * hi;

  for (int k0 = 0; k0 < KAUG; k0 += 4) {
    v2f af = *(const v2f*)(aRow + k0);
#pragma unroll
    for (int t = 0; t < 4; ++t) {
      v2f bf = *(const v2f*)(bRow + t * 16 * KAUG + k0);
      acc[t] = __builtin_amdgcn_wmma_f32_16x16x4_f32(
          false, af, false, bf, (short)0, acc[t], false, false);
    }
  }

  // Row-side parameters: acc[t][v] holds element (a = wave*16 + v + 8*hi,
  //                                               b = t*16 + laneLo)
  float d1L[8], d2L[8], pLv[8];
#pragma unroll
  for (int v = 0; v < 8; ++v) {
    int a = wave * 16 + v + 8 * hi;
    d1L[v] = dL[a];
    d2L[v] = dL[NN + a];
    pLv[v] = pL[a];
  }

  float partial = 0.f;
#pragma unroll
  for (int t = 0; t < 4; ++t) {
    int b = t * 16 + laneLo;
    float d1R = dR[b], d2R = dR[NN + b], pRv = pR[b];
#pragma unroll
    for (int v = 0; v < 8; ++v) {
      float sig = acc[t][v];
      float ntk = sig;
      ntk_layer(sig, ntk, d1L[v], d1R);
      ntk_layer(sig, ntk, d2L[v], d2R);
      partial += pLv[v] * ntk * pRv;
    }
  }

  // wave32 reduction, then cross-wave combine in LDS
  for (int off = 16; off > 0; off >>= 1)
    partial += __shfl_xor(partial, off, 32);
  __shared__ float wsum[4];
  if (lane == 0) wsum[wave] = partial;
  __syncthreads();
  if (threadIdx.x == 0)
    out[(size_t)g1 * Mright + g2] =
        (wsum[0] + wsum[1] + wsum[2] + wsum[3]) * (1.f / 4096.f);
}

// ---------------------------------------------------------------------------
// Ridge solve (64x64, Gauss-Jordan in LDS; K_SS is PSD + reg -> no pivoting)
// + prediction pred = K_ST^T sol (128x10)
// ---------------------------------------------------------------------------
__global__ void __launch_bounds__(256) solve_pred_kernel(
    const float* __restrict__ KSS, const float* __restrict__ KST,
    const float* __restrict__ yS, float* __restrict__ pred) {
  __shared__ float aug[64][75];   // 64 K cols + 10 rhs cols (+1 pad)
  __shared__ float fbuf[64];
  __shared__ float sol[64][11];
  int tid = threadIdx.x;
  for (int e = tid; e < 64 * 64; e += 256) {
    int rr = e >> 6, cc = e & 63;
    aug[rr][cc] = KSS[e];
  }
  for (int e = tid; e < 64 * 10; e += 256) {
    int rr = e / 10, cc = e - rr * 10;
    aug[rr][64 + cc] = yS[e];
  }
  __syncthreads();
  if (tid == 0) {
    float tr = 0.f;
    for (int i = 0; i < 64; ++i) tr += aug[i][i];
    float reg = 1e-6f * tr / 64.f;
    for (int i = 0; i < 64; ++i) aug[i][i] += reg;
  }
  __syncthreads();
  for (int k = 0; k < 64; ++k) {
    if (tid < 64) fbuf[tid] = (tid == k) ? 0.f : aug[tid][k] / aug[k][k];
    __syncthreads();
    for (int e = tid; e < 64 * 74; e += 256) {
      int rr = e / 74, cc = e - rr * 74;
      if (rr != k && cc >= k) aug[rr][cc] -= fbuf[rr] * aug[k][cc];
    }
    __syncthreads();
  }
  for (int e = tid; e < 640; e += 256) {
    int rr = e / 10, cc = e - rr * 10;
    sol[rr][cc] = aug[rr][64 + cc] / aug[rr][rr];
  }
  __syncthreads();
  for (int e = tid; e < 128 * 10; e += 256) {
    int t = e / 10, cc = e - t * 10;
    float a2 = 0.f;
    for (int s = 0; s < 64; ++s) a2 += KST[s * 128 + t] * sol[s][cc];
    pred[e] = a2;
  }
}

// ---------------------------------------------------------------------------
extern "C" void kernel_launch(void* const* d_in, const int* in_sizes, int n_in,
                              void* d_out, int out_size, void* d_ws, size_t ws_size,
                              hipStream_t stream) {
  const float* A_S = (const float*)d_in[0];
  const float* X_S = (const float*)d_in[1];
  const float* y_S = (const float*)d_in[2];
  const float* A_T = (const float*)d_in[3];
  const float* X_T = (const float*)d_in[4];
  float* out = (float*)d_out;          // [0,1280) = pred, [1280,5376) = K_SS
  float* ws  = (float*)d_ws;

  const int NS = 64, NT = 128;
  size_t off = 0;
  float* Y1S = ws + off; off += (size_t)NS * NN * KAUG;
  float* Y2S = ws + off; off += (size_t)NS * NN * KAUG;
  float* Y1T = ws + off; off += (size_t)NT * NN * KAUG;
  float* Y2T = ws + off; off += (size_t)NT * NN * KAUG;
  float* rS  = ws + off; off += (size_t)NS * NN;
  float* cS  = ws + off; off += (size_t)NS * NN;
  float* rT  = ws + off; off += (size_t)NT * NN;
  float* cT  = ws + off; off += (size_t)NT * NN;
  float* dS  = ws + off; off += (size_t)NS * 2 * NN;
  float* dT  = ws + off; off += (size_t)NT * 2 * NN;
  float* p1S = ws + off; off += (size_t)NS * NN;
  float* p2S = ws + off; off += (size_t)NS * NN;
  float* p1T = ws + off; off += (size_t)NT * NN;
  float* p2T = ws + off; off += (size_t)NT * NN;
  float* KST = ws + off; off += (size_t)NS * NT;

  float* KSS = out + NT * 10;  // K_SS lives directly in d_out

  prep_kernel<<<NS, 256, 0, stream>>>(A_S, X_S, Y1S, Y2S, rS, cS);
  prep_kernel<<<NT, 256, 0, stream>>>(A_T, X_T, Y1T, Y2T, rT, cT);
  postprep_kernel<<<NS, 64, 0, stream>>>(A_S, Y1S, Y2S, rS, cS, dS, p1S, p2S);
  postprep_kernel<<<NT, 64, 0, stream>>>(A_T, Y1T, Y2T, rT, cT, dT, p1T, p2T);
  pairwise_kernel<<<dim3(NS, NS), 128, 0, stream>>>(Y1S, dS, p1S, Y2S, dS, p2S, KSS, NS);
  pairwise_kernel<<<dim3(NT, NS), 128, 0, stream>>>(Y1S, dS, p1S, Y2T, dT, p2T, KST, NT);
  solve_pred_kernel<<<1, 256, 0, stream>>>(KSS, KST, y_S, out);
}